// EnsembleGRUModule2_63213328662936
// MI455X (gfx1250) — compile-verified
//
#include <hip/hip_runtime.h>
#include <hip/hip_bf16.h>
#include <math.h>

// Dims (compile-time, match reference)
#define WSTEPS 128
#define EDIM   16
#define NDIM   512   // B*I
#define FDIM   64
#define PDIM   72

typedef float v2f __attribute__((ext_vector_type(2)));
typedef float v8f __attribute__((ext_vector_type(8)));

// ---------------- Kernel A: fold (w_ih x weight_linear) -> Wc (padded to 16 gate rows), bc ----
// Wc_pad layout: [E][16][F] floats, rows g>=3 are zero (so WMMA A-fragment loads are branch-free)
// bc layout: [E][4] floats (r,z,n,pad), bc = w_ih . bias_linear + b_ih
__global__ void fold_weights_kernel(const float* __restrict__ wl,
                                    const float* __restrict__ bl,
                                    const float* __restrict__ w_ih,
                                    const float* __restrict__ b_ih,
                                    float* __restrict__ wc,
                                    float* __restrict__ bc) {
    int idx = blockIdx.x * blockDim.x + threadIdx.x;   // 16*16*64 = 16384 threads
    int e = idx >> 10;
    int g = (idx >> 6) & 15;
    int f = idx & 63;
    float acc = 0.0f;
    if (g < 3) {
        const float* wi  = w_ih + (e * 3 + g) * PDIM;
        const float* wle = wl + (size_t)e * PDIM * FDIM + f;
        #pragma unroll 8
        for (int p = 0; p < PDIM; ++p) acc += wi[p] * wle[p * FDIM];
    }
    wc[idx] = acc;
    if (f == 0 && g < 4) {
        float b = 0.0f;
        if (g < 3) {
            const float* wi  = w_ih + (e * 3 + g) * PDIM;
            const float* ble = bl + e * PDIM;
            for (int p = 0; p < PDIM; ++p) b += wi[p] * ble[p];
            b += b_ih[e * 3 + g];
        }
        bc[e * 4 + g] = b;
    }
}

// ---------------- Kernel B: gx[w,e,n,g] = sum_f Wc[e,g,f] * x[w,e,n,f]  via V_WMMA_F32_16X16X4_F32
// One wave per 16-row n-tile. A = Wc (16 gate rows x 4 f), B = x (4 f x 16 n), 16 k-steps (F=64).
// D row g, col n lives in c[g] of lane n (lanes 0-15) -> store packed float4 (r,z,n,0).
__global__ void gates_wmma_kernel(const float* __restrict__ x,
                                  const float* __restrict__ wc,
                                  float* __restrict__ gxp) {
    const int lane = threadIdx.x & 31;
    const int wave = threadIdx.x >> 5;
    const int gw   = blockIdx.x * 8 + wave;        // global wave-tile id, 65536 total
    const int w    = gw >> 9;                      // E*(N/16) = 512 tiles per timestep
    const int rem  = gw & 511;
    const int e    = rem >> 5;
    const int n0   = (rem & 31) << 4;
    const int m    = lane & 15;
    const int hi2  = (lane >> 4) << 1;             // lanes 16-31 hold K=+2,+3

    const float* ap = wc + ((e << 4) + m) * FDIM + hi2;                       // Wc_pad[e][m][*]
    const float* bp = x  + (((size_t)(w * EDIM + e)) * NDIM + (n0 + m)) * FDIM + hi2;

    v8f c = {0.f, 0.f, 0.f, 0.f, 0.f, 0.f, 0.f, 0.f};
    #pragma unroll
    for (int s = 0; s < 16; ++s) {
        v2f a = *(const v2f*)(ap + 4 * s);
        v2f b = *(const v2f*)(bp + 4 * s);
        c = __builtin_amdgcn_wmma_f32_16x16x4_f32(false, a, false, b,
                                                  (short)0, c, false, false);
    }

    if (lane < 16) {
        float4 o = make_float4(c[0], c[1], c[2], 0.0f);
        *(float4*)(gxp + (((size_t)(w * EDIM + e)) * NDIM + n0 + lane) * 4) = o;
    }
}

// ---------------- Kernel C: sequential GRU scan, one thread per (e,n) chain --------------------
__global__ void gru_scan_kernel(const float* __restrict__ gxp,
                                const float* __restrict__ bc,
                                const float* __restrict__ state,
                                const float* __restrict__ w_hh,
                                const float* __restrict__ b_hh,
                                float* __restrict__ out) {
    int idx = blockIdx.x * blockDim.x + threadIdx.x;   // E*N = 8192 threads
    int e = idx >> 9;
    float h = state[idx];                               // state[-1] flat (E,N)
    float wh0 = w_hh[e * 3 + 0], wh1 = w_hh[e * 3 + 1], wh2 = w_hh[e * 3 + 2];
    float cr  = bc[e * 4 + 0] + b_hh[e * 3 + 0];
    float cz  = bc[e * 4 + 1] + b_hh[e * 3 + 1];
    float cn  = bc[e * 4 + 2];
    float chh = b_hh[e * 3 + 2];

    const float4* gp = (const float4*)gxp + idx;        // stride per w = E*N float4s

    for (int w = 0; w < WSTEPS; ++w) {
        float4 g = gp[(size_t)w * (EDIM * NDIM)];
        float r  = 1.0f / (1.0f + __expf(-(g.x + cr + h * wh0)));
        float z  = 1.0f / (1.0f + __expf(-(g.y + cz + h * wh1)));
        float nn = tanhf(g.z + cn + r * (h * wh2 + chh));
        h = z * (h - nn) + nn;
        out[(size_t)w * (EDIM * NDIM) + idx] = h;
    }
}

extern "C" void kernel_launch(void* const* d_in, const int* in_sizes, int n_in,
                              void* d_out, int out_size, void* d_ws, size_t ws_size,
                              hipStream_t stream) {
    const float* x     = (const float*)d_in[0];   // (W,E,B,I,F)
    const float* state = (const float*)d_in[1];   // (1,E,N,1)
    const float* wl    = (const float*)d_in[2];   // (E,P,F)
    const float* bl    = (const float*)d_in[3];   // (E,P)
    const float* w_ih  = (const float*)d_in[4];   // (E,3,P)
    const float* w_hh  = (const float*)d_in[5];   // (E,3)
    const float* b_ih  = (const float*)d_in[6];   // (E,3)
    const float* b_hh  = (const float*)d_in[7];   // (E,3)
    float* out = (float*)d_out;                   // (W,E,B,I,1) fp32

    // Workspace layout (floats):
    //   [0)                      gxp: W*E*N*4     = 4,194,304   (padded gate pre-activations)
    //   [gxp)                    Wc_pad: E*16*F   = 16,384
    //   [wc)                     bc: E*4          = 64
    float* ws  = (float*)d_ws;
    float* gxp = ws;
    float* wc  = ws + (size_t)WSTEPS * EDIM * NDIM * 4;
    float* bc  = wc + EDIM * 16 * FDIM;

    // A: fold projection+gate weights (16384 threads)
    fold_weights_kernel<<<64, 256, 0, stream>>>(wl, bl, w_ih, b_ih, wc, bc);

    // B: gate pre-activations via f32 WMMA; 65536 wave-tiles, 8 waves/block
    gates_wmma_kernel<<<8192, 256, 0, stream>>>(x, wc, gxp);

    // C: GRU scan, one thread per (e,n)
    gru_scan_kernel<<<32, 256, 0, stream>>>(gxp, bc, state, w_hh, b_hh, out);
}